// CliffordLinear_27848567947545
// MI455X (gfx1250) — compile-verified
//
#include <hip/hip_runtime.h>
#include <hip/hip_bf16.h>

typedef __attribute__((ext_vector_type(2))) float v2f;
typedef __attribute__((ext_vector_type(8))) float v8f;

// Cl(3,0) structure constants: (k, p, q, sign) with T[k,p,q] = sign.
__device__ __constant__ int TERMS[64][4] = {
    {0,0,0, 1},{0,1,1, 1},{0,2,2, 1},{0,3,3, 1},{0,4,4,-1},{0,5,5,-1},{0,6,6,-1},{0,7,7,-1},
    {1,0,1, 1},{1,1,0, 1},{1,2,4,-1},{1,3,5, 1},{1,4,2, 1},{1,5,3,-1},{1,6,7,-1},{1,7,6,-1},
    {2,0,2, 1},{2,1,4, 1},{2,2,0, 1},{2,3,6,-1},{2,4,1,-1},{2,5,7, 1},{2,6,3, 1},{2,7,5,-1},
    {3,0,3, 1},{3,1,5,-1},{3,2,6, 1},{3,3,0, 1},{3,4,7,-1},{3,5,1,-1},{3,6,2,-1},{3,7,4, 1},
    {4,0,4, 1},{4,1,2, 1},{4,2,1,-1},{4,3,7, 1},{4,4,0, 1},{4,5,6,-1},{4,6,5, 1},{4,7,3,-1},
    {5,0,5, 1},{5,1,3,-1},{5,2,7, 1},{5,3,1, 1},{5,4,6, 1},{5,5,0, 1},{5,6,4,-1},{5,7,2,-1},
    {6,0,6, 1},{6,1,7,-1},{6,2,3,-1},{6,3,2, 1},{6,4,5,-1},{6,5,4, 1},{6,6,0, 1},{6,7,1, 1},
    {7,0,7, 1},{7,1,6, 1},{7,2,5,-1},{7,3,4, 1},{7,4,3, 1},{7,5,2,-1},{7,6,1, 1},{7,7,0, 1},
};

// Build W[iq][ok] = sum_p T[k,p,q] * weight[o,i,p], stored pre-swizzled into the
// WMMA f32 16x16x4 B-operand lane layout:
//   wpack[ntile*1024 + ks*64 + lane*2 + j] holds W[4*ks + 2*(lane>>4) + j][ntile*16 + (lane&15)]
__global__ void clif_pack_w(const float* __restrict__ weight, float* __restrict__ wpack) {
    int idx = blockIdx.x * blockDim.x + threadIdx.x;   // 0..4095
    if (idx >= 4096) return;
    int j     = idx & 1;
    int lane  = (idx >> 1) & 31;
    int ks    = (idx >> 6) & 15;
    int ntile = idx >> 10;
    int kk = 4 * ks + 2 * (lane >> 4) + j;   // iq row of W (0..63)
    int n  = ntile * 16 + (lane & 15);       // ok col of W (0..63)
    int i = kk >> 3, q = kk & 7;
    int o = n >> 3,  kc = n & 7;
    float acc = 0.0f;
    #pragma unroll 1
    for (int t = 0; t < 64; ++t) {
        if (TERMS[t][0] == kc && TERMS[t][2] == q) {
            acc += (float)TERMS[t][3] * weight[(o * 8 + i) * 8 + TERMS[t][1]];
        }
    }
    wpack[idx] = acc;
}

// out(B x 64) = x(B x 64) * W(64 x 64) + bias, via v_wmma_f32_16x16x4_f32.
// Block = 256 threads = 8 waves. Wave w: N-tile (w&3) of 16 cols; row-group (w>>2).
// Each block iteration covers 2 row-tiles of 16 rows; grid-stride over row-tiles.
__global__ __launch_bounds__(256)
void clif_main(const float* __restrict__ x, const float* __restrict__ wpack,
               const float* __restrict__ bias, float* __restrict__ out,
               int ntiles) {
    const int lane   = threadIdx.x & 31;
    const int wave   = threadIdx.x >> 5;
    const int ntile  = wave & 3;
    const int laneLo = lane & 15;
    const int laneHi = lane >> 4;

    // Persistent B-operand: 16 k-steps x float2 = 32 VGPRs of weights.
    v2f w[16];
    const float* wp = wpack + ntile * 1024 + lane * 2;
    #pragma unroll
    for (int ks = 0; ks < 16; ++ks) {
        w[ks] = *(const v2f*)(wp + ks * 64);
    }

    const float bv = bias[ntile * 16 + laneLo];

    const int tstride = gridDim.x * 2;
    for (int t = blockIdx.x * 2 + (wave >> 2); t < ntiles; t += tstride) {
        // A-operand: lane (laneLo) owns row t*16+laneLo; laneHi selects K pair {0,1}/{2,3}.
        const float* xrow = x + (size_t)(t * 16 + laneLo) * 64 + 2 * laneHi;
        v8f c = {};
        #pragma unroll
        for (int ks = 0; ks < 16; ++ks) {
            v2f a = *(const v2f*)(xrow + ks * 4);
            c = __builtin_amdgcn_wmma_f32_16x16x4_f32(
                    /*neg_a=*/false, a, /*neg_b=*/false, w[ks],
                    /*c_mod=*/(short)0, c, /*reuse_a=*/false, /*reuse_b=*/false);
        }
        // D layout: VGPR r, lane L -> row (r + 8*(L>>4)), col (L&15) of the 16x16 tile.
        float* orow = out + (size_t)t * 1024 + ntile * 16 + laneLo;
        #pragma unroll
        for (int r = 0; r < 8; ++r) {
            orow[(size_t)(r + 8 * laneHi) * 64] = c[r] + bv;
        }
    }
}

extern "C" void kernel_launch(void* const* d_in, const int* in_sizes, int n_in,
                              void* d_out, int out_size, void* d_ws, size_t ws_size,
                              hipStream_t stream) {
    const float* x      = (const float*)d_in[0];
    const float* weight = (const float*)d_in[1];
    const float* bias   = (const float*)d_in[2];
    float* out          = (float*)d_out;
    float* wpack        = (float*)d_ws;           // 16 KB packed weight matrix

    const int rows   = in_sizes[0] / 64;          // B = 4096*512
    const int ntiles = rows / 16;                 // 16-row tiles

    clif_pack_w<<<16, 256, 0, stream>>>(weight, wpack);

    int grid = (ntiles + 1) / 2;
    if (grid > 16384) grid = 16384;               // persistent-ish, grid-stride loop
    clif_main<<<grid, 256, 0, stream>>>(x, wpack, bias, out, ntiles);
}